// RPN_16913581211797
// MI455X (gfx1250) — compile-verified
//
#include <hip/hip_runtime.h>
#include <stdint.h>

#define NBATCH 16
#define NANCH  258048
#define PRE_K  2000
#define POST_K 1000
#define NMS_TH 0.7f
#define SCALE_CLAMP_F 3.3322045101752038f   // log(224/8)
#define BGV (-1e8f)
#define HPAD 257                             // bank-decorrelating pad for wave-private hists

typedef unsigned int u32x4 __attribute__((ext_vector_type(4)));
typedef int          i32x4 __attribute__((ext_vector_type(4)));
typedef int          i32x8 __attribute__((ext_vector_type(8)));

#if defined(__has_builtin)
#  if __has_builtin(__builtin_amdgcn_tensor_load_to_lds) && __has_builtin(__builtin_amdgcn_s_wait_tensorcnt)
#    define USE_TDM 1
#  endif
#endif

#if defined(USE_TDM)
// Issue one 1-D TDM copy: nElem f32 from global gaddr -> LDS lds_off.
__device__ __forceinline__ void tdm_load_f32(unsigned lds_off,
                                             unsigned long long ga,
                                             int nElem) {
  u32x4 g0;
  g0.x = 1u;                                           // count=1, user mode
  g0.y = lds_off;                                      // lds_addr
  g0.z = (unsigned)(ga & 0xFFFFFFFFull);               // global_addr[31:0]
  g0.w = (unsigned)((ga >> 32) & 0x01FFFFFFull) | 0x80000000u; // addr[56:32]|type=2
  i32x8 g1;
  g1[0] = 0x00020000;                                  // wg_mask=0, data_size=4B
  g1[1] = (nElem & 0xFFFF) << 16;                      // tensor_dim0 low16 (bits 79:48)
  g1[2] = ((nElem >> 16) & 0xFFFF) | (1 << 16);        // tensor_dim0 hi16, tensor_dim1=1
  g1[3] = (nElem & 0xFFFF) << 16;                      // tile_dim0 (<=65535)
  g1[4] = 1;                                           // tile_dim1=1, tile_dim2=0
  g1[5] = nElem;                                       // tensor_dim0_stride
  g1[6] = 0;
  g1[7] = 0;
  i32x4 gz4 = {0, 0, 0, 0};
  i32x8 gz8 = {0, 0, 0, 0, 0, 0, 0, 0};
  __builtin_amdgcn_tensor_load_to_lds(g0, g1, gz4, gz4, gz8, 0);
}
#endif

// ---------------------------------------------------------------------------
// Kernel 1: per-batch exact top-2000 via 4-pass (8-bit digit) radix select
// with wave-private (bank-padded) histograms, then threshold compaction.
// One workgroup (16 waves of 32) per batch.
// ---------------------------------------------------------------------------
__global__ __launch_bounds__(512) void topk_radix_select(
    const float* __restrict__ scores, int* __restrict__ sel_idx,
    float* __restrict__ sel_sc) {
  const int b = blockIdx.x;
  const float* sp = scores + (size_t)b * NANCH;
  __shared__ unsigned whist[16 * HPAD];   // wave-private histograms (padded)
  __shared__ unsigned hist[256];          // reduced histogram
  __shared__ unsigned s_prefix;
  __shared__ unsigned s_rem;
  __shared__ unsigned s_pos, s_eq;
  const int tid  = threadIdx.x;
  const int wave = tid >> 5;

  if (tid == 0) { s_prefix = 0u; s_rem = PRE_K; }

  for (int pass = 0; pass < 4; ++pass) {
    const int shift = 24 - 8 * pass;
    for (int i = tid; i < 16 * HPAD; i += 512) whist[i] = 0u;
    __syncthreads();
    const unsigned pmask = pass ? (0xFFFFFFFFu << (shift + 8)) : 0u;
    const unsigned pref  = s_prefix;
    unsigned* myh = &whist[wave * HPAD];
    for (int i = tid; i < NANCH; i += 512) {
      __builtin_prefetch(sp + i + 512 * 8, 0, 1);   // global_prefetch_b8
      unsigned u = __float_as_uint(sp[i]);
      unsigned k = (u & 0x80000000u) ? ~u : (u | 0x80000000u);
      if ((k & pmask) == pref)
        atomicAdd(&myh[(k >> shift) & 0xFFu], 1u);
    }
    __syncthreads();
    if (tid < 256) {                       // reduce 16 private copies
      unsigned s = 0;
      #pragma unroll
      for (int w = 0; w < 16; ++w) s += whist[w * HPAD + tid];
      hist[tid] = s;
    }
    __syncthreads();
    if (tid == 0) {
      unsigned rem = s_rem, cum = 0; int sel = 0;
      for (int d = 255; d >= 0; --d) {
        unsigned c = hist[d];
        if (cum + c >= rem) { sel = d; break; }
        cum += c;
      }
      s_prefix = pref | ((unsigned)sel << shift);
      s_rem = rem - cum;
    }
    __syncthreads();
  }

  const unsigned T = s_prefix;   // exact 2000th-largest key
  const unsigned r = s_rem;      // how many == T to take
  if (tid == 0) { s_pos = 0u; s_eq = 0u; }
  __syncthreads();

  for (int i = tid; i < NANCH; i += 512) {
    __builtin_prefetch(sp + i + 512 * 8, 0, 1);
    float v = sp[i];
    unsigned u = __float_as_uint(v);
    unsigned k = (u & 0x80000000u) ? ~u : (u | 0x80000000u);
    if (k > T) {
      unsigned p = atomicAdd(&s_pos, 1u);
      sel_idx[b * PRE_K + p] = i;
      sel_sc [b * PRE_K + p] = v;
    } else if (k == T) {
      unsigned q = atomicAdd(&s_eq, 1u);
      if (q < r) {
        unsigned p = atomicAdd(&s_pos, 1u);
        sel_idx[b * PRE_K + p] = i;
        sel_sc [b * PRE_K + p] = v;
      }
    }
  }
}

// ---------------------------------------------------------------------------
// Kernel 2: gather only the 16x2000 winning deltas, reconstruct the anchor
// analytically from the flat index, apply deltas, clip.  Boxes are written
// SoA per batch: sel_boxes[b*8000 + plane*2000 + j], plane = x1,y1,x2,y2 —
// a contiguous 8000-f32 block per batch (one TDM descriptor, LDS
// conflict-free in the NMS sweep).
// ---------------------------------------------------------------------------
__global__ __launch_bounds__(256) void gather_decode(
    const float* __restrict__ deltas, const int* __restrict__ sel_idx,
    float* __restrict__ sel_boxes) {
  int t = blockIdx.x * 256 + threadIdx.x;
  if (t >= NBATCH * PRE_K) return;
  int b = t / PRE_K;
  int j = t - b * PRE_K;
  int idx = sel_idx[t];

  int stride, base;
  if (idx < 196608)      { stride = 8;  base = 0;      }
  else if (idx < 245760) { stride = 16; base = 196608; }
  else                   { stride = 32; base = 245760; }
  int n   = 2048 / stride;
  int rel = idx - base;
  int loc = rel / 3;
  int arq = rel - loc * 3;                      // aspect-ratio slot 0/1/2
  float cx = ((float)(loc % n) + 0.5f) * (float)stride;
  float cy = ((float)(loc / n) + 0.5f) * (float)stride;
  float side = (float)(stride * 8);             // STRIDE_SCALE = 8
  float area = side * side;
  float arv  = (arq == 0) ? 0.5f : ((arq == 1) ? 1.0f : 2.0f);
  float w = sqrtf(area / arv);
  float h = area / w;
  float ax1 = cx - w * 0.5f, ay1 = cy - h * 0.5f;
  float ax2 = ax1 + w,       ay2 = ay1 + h;
  float px = (ax1 + ax2) * 0.5f, py = (ay1 + ay2) * 0.5f;
  float pw = ax2 - ax1,          ph = ay2 - ay1;

  const float* dp = deltas + ((size_t)b * NANCH + (size_t)idx) * 4;
  float4 d = *(const float4*)dp;

  float o0, o1, o2, o3;
  if (d.x != BGV) {
    float dw = fminf(d.z, SCALE_CLAMP_F);
    float dh = fminf(d.w, SCALE_CLAMP_F);
    float bx = d.x * pw + px;
    float by = d.y * ph + py;
    float bw = expf(dw) * pw;
    float bh = expf(dh) * ph;
    o0 = fminf(fmaxf(bx - bw * 0.5f, 0.f), 2048.f);
    o1 = fminf(fmaxf(by - bh * 0.5f, 0.f), 2048.f);
    o2 = fminf(fmaxf(bx + bw * 0.5f, 0.f), 2048.f);
    o3 = fminf(fmaxf(by + bh * 0.5f, 0.f), 2048.f);
  } else {
    o0 = o1 = o2 = o3 = BGV;
  }
  float* ob = sel_boxes + (size_t)b * (PRE_K * 4);
  ob[0 * PRE_K + j] = o0;
  ob[1 * PRE_K + j] = o1;
  ob[2 * PRE_K + j] = o2;
  ob[3 * PRE_K + j] = o3;
}

// ---------------------------------------------------------------------------
// Kernel 3: greedy NMS, one WG (32 waves of 32) per batch; SoA boxes and
// scores staged into LDS via the Tensor Data Mover.
// ---------------------------------------------------------------------------
__global__ __launch_bounds__(1024) void nms_kernel(
    const float* __restrict__ sel_boxes, const float* __restrict__ sel_sc,
    float* __restrict__ out) {
  const int b    = blockIdx.x;
  const int tid  = threadIdx.x;
  const int wave = tid >> 5;
  const int lane = tid & 31;

  __shared__ float bxs[PRE_K * 4];   // SoA: [x1 | y1 | x2 | y2], 32000 B
  __shared__ float sm [PRE_K];       // masked scores
  __shared__ float ara[PRE_K];       // areas
  __shared__ float wbest[32];
  __shared__ int   wbi[32];
  __shared__ float s_piv[5];
  __shared__ int   s_stop;

  float* X1 = &bxs[0 * PRE_K];
  float* Y1 = &bxs[1 * PRE_K];
  float* X2 = &bxs[2 * PRE_K];
  float* Y2 = &bxs[3 * PRE_K];

  const float* gb = sel_boxes + (size_t)b * (PRE_K * 4);
  const float* gs = sel_sc + (size_t)b * PRE_K;

#if defined(USE_TDM)
  if (wave == 0) {
    tdm_load_f32((unsigned)(uintptr_t)&bxs[0],
                 (unsigned long long)(uintptr_t)gb, PRE_K * 4);
    tdm_load_f32((unsigned)(uintptr_t)&sm[0],
                 (unsigned long long)(uintptr_t)gs, PRE_K);
    __builtin_amdgcn_s_wait_tensorcnt((short)0);
  }
#else
  for (int i = tid; i < PRE_K * 4; i += 1024) bxs[i] = gb[i];
  for (int i = tid; i < PRE_K; i += 1024) sm[i] = gs[i];
#endif
  __syncthreads();
  for (int i = tid; i < PRE_K; i += 1024)
    ara[i] = (X2[i] - X1[i]) * (Y2[i] - Y1[i]);
  __syncthreads();

  float* out_boxes = out + (size_t)b * POST_K * 4;
  float* out_sc    = out + (size_t)NBATCH * POST_K * 4 + (size_t)b * POST_K;
  const float NEG = -__builtin_inff();

  int it = 0;
  for (; it < POST_K; ++it) {
    // --- argmax over sm[0..1999]: per-thread -> wave32 shuffle -> wave0 ---
    float best = NEG; int bi = 0x7FFFFFFF;
    for (int i = tid; i < PRE_K; i += 1024) {
      float v = sm[i];
      if (v > best) { best = v; bi = i; }
    }
    for (int off = 16; off > 0; off >>= 1) {
      float ov = __shfl_down(best, (unsigned)off, 32);
      int   oi = __shfl_down(bi,   (unsigned)off, 32);
      if (ov > best || (ov == best && oi < bi)) { best = ov; bi = oi; }
    }
    if (lane == 0) { wbest[wave] = best; wbi[wave] = bi; }
    __syncthreads();
    if (wave == 0) {
      best = wbest[lane]; bi = wbi[lane];
      for (int off = 16; off > 0; off >>= 1) {
        float ov = __shfl_down(best, (unsigned)off, 32);
        int   oi = __shfl_down(bi,   (unsigned)off, 32);
        if (ov > best || (ov == best && oi < bi)) { best = ov; bi = oi; }
      }
      if (lane == 0) {
        if (best > NEG) {
          float p0 = X1[bi], p1 = Y1[bi], p2 = X2[bi], p3 = Y2[bi];
          s_piv[0] = p0; s_piv[1] = p1; s_piv[2] = p2; s_piv[3] = p3;
          s_piv[4] = ara[bi];
          s_stop = 0;
          out_boxes[it * 4 + 0] = p0;
          out_boxes[it * 4 + 1] = p1;
          out_boxes[it * 4 + 2] = p2;
          out_boxes[it * 4 + 3] = p3;
          out_sc[it] = best;
        } else {
          s_stop = 1;
        }
      }
    }
    __syncthreads();
    if (s_stop) break;               // remaining slots filled below

    // --- suppression sweep (matches reference: keep iff iou < thresh) ---
    float p0 = s_piv[0], p1 = s_piv[1], p2 = s_piv[2], p3 = s_piv[3];
    float pa = s_piv[4];
    for (int i = tid; i < PRE_K; i += 1024) {
      float xx1 = fmaxf(p0, X1[i]);
      float yy1 = fmaxf(p1, Y1[i]);
      float xx2 = fminf(p2, X2[i]);
      float yy2 = fminf(p3, Y2[i]);
      float inter = fmaxf(xx2 - xx1, 0.f) * fmaxf(yy2 - yy1, 0.f);
      float iou = inter / (pa + ara[i] - inter);
      if (!(iou < NMS_TH)) sm[i] = NEG;   // NaN suppresses, like reference
    }
    __syncthreads();
  }

  // tail: zeros / -inf for exhausted iterations (reference ok=false path)
  for (int j = it + tid; j < POST_K; j += 1024) {
    out_boxes[j * 4 + 0] = 0.f;
    out_boxes[j * 4 + 1] = 0.f;
    out_boxes[j * 4 + 2] = 0.f;
    out_boxes[j * 4 + 3] = 0.f;
    out_sc[j] = NEG;
  }
}

// ---------------------------------------------------------------------------
extern "C" void kernel_launch(void* const* d_in, const int* in_sizes, int n_in,
                              void* d_out, int out_size, void* d_ws, size_t ws_size,
                              hipStream_t stream) {
  (void)in_sizes; (void)n_in; (void)out_size; (void)ws_size;
  const float* deltas = (const float*)d_in[3];   // (16, 258048, 4)
  const float* scores = (const float*)d_in[4];   // (16, 258048)

  int*   sel_idx   = (int*)d_ws;                                 // 32000 ints
  float* sel_sc    = (float*)((char*)d_ws + 128 * 1024);         // 32000 f32
  float* sel_boxes = (float*)((char*)d_ws + 256 * 1024);         // 128000 f32 (SoA per batch)

  topk_radix_select<<<NBATCH, 512, 0, stream>>>(scores, sel_idx, sel_sc);
  gather_decode<<<(NBATCH * PRE_K + 255) / 256, 256, 0, stream>>>(deltas, sel_idx, sel_boxes);
  nms_kernel<<<NBATCH, 1024, 0, stream>>>(sel_boxes, sel_sc, (float*)d_out);
}